// DlrmSmall_41506563948779
// MI455X (gfx1250) — compile-verified
//
#include <hip/hip_runtime.h>
#include <hip/hip_bf16.h>

typedef __attribute__((ext_vector_type(2))) float v2f;
typedef __attribute__((ext_vector_type(8))) float v8f;

#define TM 128      // block tile M
#define TN 64       // block tile N
#define TK 32       // K step
#define LDA_S 36    // padded LDS stride for A tile (mult of 4 -> aligned float4 stores; m*36%64 distinct -> conflict-free frags)
#define LDB_S 72    // padded LDS stride for B tile (2*72 % 64 == 16 -> half-waves hit disjoint banks; mult of 4)

// Generic fp32 GEMM: C[M,N] = act(A[M,K] (row-major, leading dim lda) @ B[K,N] (row-major, ld=N) + bias[N])
// Block: 256 threads = 8 waves. Wave computes a 32x32 C tile via 2x2 v_wmma_f32_16x16x4_f32 accumulators.
// If lda % 4 == 0, A-tile staging is unguarded float4 (requires: columns [K, roundup32(K)) of A readable and
// harmless — we zero-pad the top-input buffer so products against the zeroed B tile rows are exact zeros).
__global__ __launch_bounds__(256) void gemm_f32_wmma(
    const float* __restrict__ A, const float* __restrict__ B,
    const float* __restrict__ bias, float* __restrict__ C,
    int M, int N, int K, int lda, int relu)
{
    __shared__ float sA[TM * LDA_S];
    __shared__ float sB[TK * LDB_S];

    const int tid  = threadIdx.x;
    const int m0   = blockIdx.y * TM;
    const int n0   = blockIdx.x * TN;
    const int wave = tid >> 5;
    const int lane = tid & 31;
    const int wm   = wave & 3;   // 4 M sub-tiles of 32 rows
    const int wn   = wave >> 2;  // 2 N sub-tiles of 32 cols
    const int lm   = lane & 15;
    const int lk2  = (lane >> 4) * 2;  // upper half-wave holds K+2,K+3

    const bool veca = (lda & 3) == 0;

    v8f acc[2][2] = {};

    for (int kt = 0; kt < K; kt += TK) {
        // --- stage A tile (TM x TK) ---
        if (veca) {
            // float4 path: 128 rows x 8 vec4 = 1024 vec loads / 256 threads
            for (int i = tid; i < TM * (TK / 4); i += 256) {
                int r = i >> 3, c4 = (i & 7) * 4;
                float4 v = *(const float4*)(A + (size_t)(m0 + r) * lda + kt + c4);
                *(float4*)(&sA[r * LDA_S + c4]) = v;
            }
        } else {
            for (int i = tid; i < TM * TK; i += 256) {
                int r = i >> 5, c = i & 31;
                int k = kt + c;
                sA[r * LDA_S + c] = (k < K) ? A[(size_t)(m0 + r) * lda + k] : 0.0f;
            }
        }
        // --- stage B tile (TK x TN), zero rows past K (N always mult of 64, 16B-aligned) ---
        for (int i = tid; i < TK * (TN / 4); i += 256) {
            int r = i >> 4, c4 = (i & 15) * 4;
            int k = kt + r;
            float4 v = make_float4(0.f, 0.f, 0.f, 0.f);
            if (k < K) v = *(const float4*)(B + (size_t)k * N + n0 + c4);
            *(float4*)(&sB[r * LDB_S + c4]) = v;
        }
        __syncthreads();

        #pragma unroll
        for (int kk = 0; kk < TK; kk += 4) {
            // A fragments: lane holds rows (base + lane%16), K = kk + 2*(lane/16) + {0,1}
            const float* pa = &sA[(wm * 32 + lm) * LDA_S + kk + lk2];
            v2f a0 = *(const v2f*)pa;
            v2f a1 = *(const v2f*)(pa + 16 * LDA_S);
            // B fragments: lane holds col (base + lane%16), rows kk+2*(lane/16) and +1
            int kb = kk + lk2;
            int cb = wn * 32 + lm;
            v2f b0, b1;
            b0.x = sB[kb * LDB_S + cb];
            b0.y = sB[(kb + 1) * LDB_S + cb];
            b1.x = sB[kb * LDB_S + cb + 16];
            b1.y = sB[(kb + 1) * LDB_S + cb + 16];

            acc[0][0] = __builtin_amdgcn_wmma_f32_16x16x4_f32(false, a0, false, b0, (short)0, acc[0][0], false, false);
            acc[0][1] = __builtin_amdgcn_wmma_f32_16x16x4_f32(false, a0, false, b1, (short)0, acc[0][1], false, false);
            acc[1][0] = __builtin_amdgcn_wmma_f32_16x16x4_f32(false, a1, false, b0, (short)0, acc[1][0], false, false);
            acc[1][1] = __builtin_amdgcn_wmma_f32_16x16x4_f32(false, a1, false, b1, (short)0, acc[1][1], false, false);
        }
        __syncthreads();
    }

    // Epilogue: C/D layout -> VGPR r: rows r (lanes 0-15) and r+8 (lanes 16-31); col = lane%16
    const int lhalf = (lane >> 4) * 8;
    #pragma unroll
    for (int X = 0; X < 2; ++X) {
        const int rowb = m0 + wm * 32 + X * 16 + lhalf;
        #pragma unroll
        for (int Y = 0; Y < 2; ++Y) {
            const int col = n0 + wn * 32 + Y * 16 + lm;
            const float bv = bias[col];
            #pragma unroll
            for (int r = 0; r < 8; ++r) {
                float v = acc[X][Y][r] + bv;
                if (relu) v = fmaxf(v, 0.0f);
                C[(size_t)(rowb + r) * N + col] = v;
            }
        }
    }
}

#define NSPARSE 26
#define NFEAT 27
#define EMBD 128
#define FPAD 132      // padded feature stride in LDS (mult of 4 for float4; 132%64 spreads banks across rows)
#define VOCAB_SZ 100000
#define TOPSTRIDE 512 // top-input row stride (506 used + 6 zeroed pad, so GEMM can vector-load unguarded)

// One block per batch row: gather 26 embeddings + bot_out into LDS (float4 streams), compute 27x27 Gram
// upper triangle (378 dots of length 128), emit [bot_out(128) | triu(378) | 0 pad(6)].
__global__ __launch_bounds__(256) void interact_kernel(
    const float* __restrict__ bot, const int* __restrict__ cat,
    const float* __restrict__ emb, float* __restrict__ top)
{
    __shared__ float sF[NFEAT * FPAD];
    const int row = blockIdx.x;
    const int tid = threadIdx.x;

    // vectorized gather: 27 rows x 32 float4
    for (int i = tid; i < NFEAT * (EMBD / 4); i += 256) {
        int f = i >> 5, c4 = (i & 31) * 4;
        float4 v;
        if (f == 0) {
            v = *(const float4*)(bot + (size_t)row * EMBD + c4);
        } else {
            int id = cat[row * NSPARSE + (f - 1)] + (f - 1) * VOCAB_SZ;
            v = *(const float4*)(emb + (size_t)id * EMBD + c4);
        }
        *(float4*)(&sF[f * FPAD + c4]) = v;
    }
    __syncthreads();

    const size_t ob = (size_t)row * TOPSTRIDE;
    for (int c = tid; c < EMBD; c += 256) top[ob + c] = sF[c];

    for (int p = tid; p < 378; p += 256) {
        // invert row-major upper-triangle (incl. diagonal) linear index
        int i = 0, rem = p;
        while (rem >= NFEAT - i) { rem -= NFEAT - i; ++i; }
        int j = i + rem;
        const float4* fi = (const float4*)&sF[i * FPAD];
        const float4* fj = (const float4*)&sF[j * FPAD];
        float s = 0.0f;
        #pragma unroll 4
        for (int k = 0; k < EMBD / 4; ++k) {
            float4 a = fi[k], b = fj[k];
            s += a.x * b.x + a.y * b.y + a.z * b.z + a.w * b.w;
        }
        top[ob + EMBD + p] = s;
    }
    // zero the 6-column pad so downstream GEMM float4 A-loads read benign zeros
    if (tid < TOPSTRIDE - (EMBD + 378)) top[ob + EMBD + 378 + tid] = 0.0f;
}

// Final 256 -> 1 layer: one wave per row, lane-strided float4 dot + wave32 butterfly reduce. No ReLU.
__global__ __launch_bounds__(256) void final_layer(
    const float* __restrict__ A, const float* __restrict__ w,
    const float* __restrict__ bias, float* __restrict__ out)
{
    const int wave = threadIdx.x >> 5;
    const int lane = threadIdx.x & 31;
    const int row  = blockIdx.x * 8 + wave;
    const float4* a = (const float4*)(A + (size_t)row * 256 + lane * 8);
    const float4* ww = (const float4*)(w + lane * 8);
    float s = 0.0f;
    #pragma unroll
    for (int k = 0; k < 2; ++k) {
        float4 x = a[k], y = ww[k];
        s += x.x * y.x + x.y * y.y + x.z * y.z + x.w * y.w;
    }
    #pragma unroll
    for (int off = 16; off > 0; off >>= 1) s += __shfl_xor(s, off, 32);
    if (lane == 0) out[row] = s + bias[0];
}

extern "C" void kernel_launch(void* const* d_in, const int* in_sizes, int n_in,
                              void* d_out, int out_size, void* d_ws, size_t ws_size,
                              hipStream_t stream) {
    const int B = 8192;
    // input order: dense, cat, bot_ws[0..2], bot_bs[0..2], emb, top_ws[0..4], top_bs[0..4]
    const float* dense = (const float*)d_in[0];
    const int*   cat   = (const int*)d_in[1];
    const float* bw0   = (const float*)d_in[2];
    const float* bw1   = (const float*)d_in[3];
    const float* bw2   = (const float*)d_in[4];
    const float* bb0   = (const float*)d_in[5];
    const float* bb1   = (const float*)d_in[6];
    const float* bb2   = (const float*)d_in[7];
    const float* emb   = (const float*)d_in[8];
    const float* tw0   = (const float*)d_in[9];
    const float* tw1   = (const float*)d_in[10];
    const float* tw2   = (const float*)d_in[11];
    const float* tw3   = (const float*)d_in[12];
    const float* tw4   = (const float*)d_in[13];
    const float* tb0   = (const float*)d_in[14];
    const float* tb1   = (const float*)d_in[15];
    const float* tb2   = (const float*)d_in[16];
    const float* tb3   = (const float*)d_in[17];
    const float* tb4   = (const float*)d_in[18];
    float* out = (float*)d_out;

    // workspace: two ping-pong 8192x1024 buffers + 8192x128 bot_out
    float* bufA = (float*)d_ws;
    float* bufB = bufA + (size_t)B * 1024;
    float* botO = bufB + (size_t)B * 1024;

    dim3 blk(256);
    auto grid = [&](int N, int M) { return dim3(N / TN, M / TM); };

    // bottom MLP (first layer: lda=13 -> guarded scalar staging path)
    gemm_f32_wmma<<<grid(512, B), blk, 0, stream>>>(dense, bw0, bb0, bufA, B, 512, 13, 13, 1);
    gemm_f32_wmma<<<grid(256, B), blk, 0, stream>>>(bufA, bw1, bb1, bufB, B, 256, 512, 512, 1);
    gemm_f32_wmma<<<grid(128, B), blk, 0, stream>>>(bufB, bw2, bb2, botO, B, 128, 256, 256, 1);

    // embedding gather + pairwise interaction -> top input (padded ld=512, pad zeroed)
    interact_kernel<<<dim3(B), blk, 0, stream>>>(botO, cat, emb, bufA);

    // top MLP
    gemm_f32_wmma<<<grid(1024, B), blk, 0, stream>>>(bufA, tw0, tb0, bufB, B, 1024, 506, TOPSTRIDE, 1);
    gemm_f32_wmma<<<grid(1024, B), blk, 0, stream>>>(bufB, tw1, tb1, bufA, B, 1024, 1024, 1024, 1);
    gemm_f32_wmma<<<grid(512, B),  blk, 0, stream>>>(bufA, tw2, tb2, bufB, B, 512, 1024, 1024, 1);
    gemm_f32_wmma<<<grid(256, B),  blk, 0, stream>>>(bufB, tw3, tb3, bufA, B, 256, 512, 512, 1);

    // final 256 -> 1 (no ReLU)
    final_layer<<<dim3(B / 8), blk, 0, stream>>>(bufA, tw4, tb4, out);
}